// _ProposalLayer_9809705304714
// MI455X (gfx1250) — compile-verified
//
#include <hip/hip_runtime.h>
#include <cstdint>
#include <cstddef>

// ---------------- problem constants (match reference setup_inputs) ----------
#define BATCH   16
#define NANCH   9
#define HH      128
#define WW      128
#define HW      (HH * WW)          // 16384
#define NPROP   (HW * NANCH)       // 147456 proposals per batch
#define PRE_NMS 6000
#define POST_NMS 300
#define NMS_TH  0.7f
#define SORT_N  8192               // next pow2 >= PRE_NMS
#define NMS_CHUNK 3000             // boxes staged per TDM chunk in NMS

// ---------------- workspace layout (bytes, all 64-aligned) -----------------
static constexpr size_t OFF_PROP = 0;                                   // BATCH*NPROP*16 = 37,748,736
static constexpr size_t OFF_KEYS = OFF_PROP + (size_t)BATCH * NPROP * 16;   // +18,874,368
static constexpr size_t OFF_TOPK = OFF_KEYS + (size_t)BATCH * NPROP * 8;    // +768,000
static constexpr size_t OFF_SBOX = OFF_TOPK + (size_t)BATCH * PRE_NMS * 8;  // +1,536,000
static constexpr size_t OFF_HIST = OFF_SBOX + (size_t)BATCH * PRE_NMS * 16; // +16,384
static constexpr size_t OFF_PREF = OFF_HIST + (size_t)BATCH * 256 * 4;      // +128
static constexpr size_t OFF_NEED = OFF_PREF + (size_t)BATCH * 8;            // +64
static constexpr size_t OFF_CNT  = OFF_NEED + 64;                           // +64

// ---------------- Tensor Data Mover: 1-D u64 tile copy, global -> LDS -------
#if defined(__gfx1250__) && __has_builtin(__builtin_amdgcn_tensor_load_to_lds)
#define HAVE_TDM 1
#else
#define HAVE_TDM 0
#endif

// ---------------- WMMA (f32 16x16x4) availability ---------------------------
#if defined(__gfx1250__) && __has_builtin(__builtin_amdgcn_wmma_f32_16x16x4_f32)
#define HAVE_WMMA4 1
#else
#define HAVE_WMMA4 0
#endif

typedef unsigned int tdm_u32x4 __attribute__((ext_vector_type(4)));
typedef int          tdm_i32x4 __attribute__((ext_vector_type(4)));
typedef int          tdm_i32x8 __attribute__((ext_vector_type(8)));
typedef float        v2f       __attribute__((ext_vector_type(2)));
typedef float        v8f       __attribute__((ext_vector_type(8)));

#if HAVE_TDM
// Issue a TDM DMA of n64 contiguous 8-byte elements from global to LDS and
// wait for completion (TENSORcnt==0). Call from a single wave (e.g. tid==0).
// D# built per CDNA5 ISA (count=1, type=2, data_size=8B,
// 1-D tile: tile_dim0 = n64 <= 65535, tile_dim1 = 1, no pad/iterate/multicast).
__device__ __forceinline__ void tdm_load_1d_u64(const void* gsrc, void* lds_dst,
                                                unsigned n64) {
    uint64_t ga  = (uint64_t)(uintptr_t)gsrc;
    uint32_t lds = (uint32_t)(uintptr_t)lds_dst;   // low 32b of generic = LDS offset
    tdm_u32x4 g0;
    g0[0] = 1u;                                    // count=1 valid, user mode
    g0[1] = lds;                                   // lds_addr
    g0[2] = (uint32_t)(ga & 0xFFFFFFFFull);        // global_addr[31:0]
    g0[3] = (uint32_t)((ga >> 32) & 0x01FFFFFFull) | 0x80000000u; // [56:32] | type=2
    tdm_i32x8 g1;
    g1[0] = (int)(3u << 16);                       // data_size=3 (8 bytes)
    g1[1] = (int)((n64 & 0xFFFFu) << 16);          // tensor_dim0[15:0]
    g1[2] = (int)((n64 >> 16) | (1u << 16));       // tensor_dim0 hi | tensor_dim1=1
    g1[3] = (int)(n64 << 16);                      // tile_dim0 = n64
    g1[4] = (int)1;                                // tile_dim1=1, tile_dim2=0
    g1[5] = (int)n64;                              // tensor_dim0_stride lo
    g1[6] = (int)(n64 << 16);                      // stride hi=0 | dim1_stride lo
    g1[7] = 0;                                     // dim1_stride hi
    tdm_i32x4 z4 = {0, 0, 0, 0};
#if defined(__clang_major__) && (__clang_major__ >= 23)
    tdm_i32x8 z8 = {0, 0, 0, 0, 0, 0, 0, 0};
    __builtin_amdgcn_tensor_load_to_lds(g0, g1, z4, z4, z8, 0);
#else
    __builtin_amdgcn_tensor_load_to_lds(g0, g1, z4, z4, 0);
#endif
#if __has_builtin(__builtin_amdgcn_s_wait_tensorcnt)
    __builtin_amdgcn_s_wait_tensorcnt(0);
#endif
}
#endif // HAVE_TDM

// ============================================================================
// Kernel 0: init select state
// ============================================================================
__global__ void k_init(unsigned* hist, unsigned long long* pref,
                       unsigned* need, unsigned* cnt) {
    int b = blockIdx.x;
    hist[b * 256 + threadIdx.x] = 0u;
    if (threadIdx.x == 0) {
        pref[b] = 0ull;
        need[b] = PRE_NMS;
        cnt[b]  = 0u;
    }
}

// ============================================================================
// Kernel 1: decode boxes, clip, build 64-bit sort keys
// key = (orderedScoreBits << 32) | (0xFFFFFFFF - proposalIdx)
// descending key order == descending score, ascending index (distinct keys)
// ============================================================================
__global__ void k_transform(const float* __restrict__ scores_map,
                            const float* __restrict__ deltas,
                            const float* __restrict__ im_info,
                            const float* __restrict__ anchors,
                            float4* __restrict__ prop,
                            unsigned long long* __restrict__ keys) {
    int t = blockIdx.x * blockDim.x + threadIdx.x;
    if (t >= BATCH * NANCH * HW) return;
    int pos = t % HW;                   // h*W + w  (fastest -> coalesced)
    int a   = (t / HW) % NANCH;
    int b   = t / (HW * NANCH);

    size_t sidx  = ((size_t)b * (2 * NANCH) + NANCH + a) * HW + pos;
    float  s     = scores_map[sidx];
    size_t dbase = ((size_t)b * (4 * NANCH) + 4 * a) * HW + pos;
    float dx = deltas[dbase];
    float dy = deltas[dbase + HW];
    float dw = deltas[dbase + 2 * HW];
    float dh = deltas[dbase + 3 * HW];
    if (a + 1 < NANCH) {                // CDNA5 speculative prefetch of next anchor's deltas
        __builtin_prefetch(deltas + dbase + 4 * (size_t)HW, 0, 1);
    }

    float ax1 = anchors[a * 4 + 0], ay1 = anchors[a * 4 + 1];
    float ax2 = anchors[a * 4 + 2], ay2 = anchors[a * 4 + 3];
    float sx = (float)((pos & (WW - 1)) * 16);
    float sy = (float)((pos >> 7) * 16);
    float x1 = ax1 + sx, y1 = ay1 + sy, x2 = ax2 + sx, y2 = ay2 + sy;

    float w  = x2 - x1 + 1.0f;
    float h  = y2 - y1 + 1.0f;
    float cx = x1 + 0.5f * w;
    float cy = y1 + 0.5f * h;
    float pcx = dx * w + cx;
    float pcy = dy * h + cy;
    float pw  = expf(dw) * w;
    float ph  = expf(dh) * h;
    float px1 = pcx - 0.5f * pw, py1 = pcy - 0.5f * ph;
    float px2 = pcx + 0.5f * pw, py2 = pcy + 0.5f * ph;

    float limx = im_info[b * 3 + 1] - 1.0f;
    float limy = im_info[b * 3 + 0] - 1.0f;
    px1 = fminf(fmaxf(px1, 0.0f), limx);
    py1 = fminf(fmaxf(py1, 0.0f), limy);
    px2 = fminf(fmaxf(px2, 0.0f), limx);
    py2 = fminf(fmaxf(py2, 0.0f), limy);

    int k = pos * NANCH + a;            // reference flatten order (h,w,a)
    size_t o = (size_t)b * NPROP + k;
    prop[o] = make_float4(px1, py1, px2, py2);

    unsigned u = __float_as_uint(s);
    u = (u & 0x80000000u) ? ~u : (u | 0x80000000u);   // monotonic with float value
    keys[o] = ((unsigned long long)u << 32) |
              (unsigned long long)(0xFFFFFFFFu - (unsigned)k);
}

// ============================================================================
// Kernel 2a/2b: exact 64-bit radix select (8 x 8-bit digits, MSB first)
// Finds T[b] = 6000th-largest key; keys distinct => exactly 6000 keys >= T.
// ============================================================================
__global__ void k_hist(const unsigned long long* __restrict__ keys,
                       const unsigned long long* __restrict__ pref,
                       unsigned* __restrict__ hist, int round) {
    __shared__ unsigned lh[256];
    for (int i = threadIdx.x; i < 256; i += blockDim.x) lh[i] = 0u;
    __syncthreads();
    int t = blockIdx.x * blockDim.x + threadIdx.x;   // NPROP % 256 == 0 -> b uniform
    int b = t / NPROP;
    unsigned long long key  = keys[t];
    unsigned long long p    = pref[b];
    unsigned long long mask = (round == 0) ? 0ull : (~0ull) << (64 - 8 * round);
    if ((key & mask) == p) {
        unsigned digit = (unsigned)((key >> (56 - 8 * round)) & 0xFFull);
        atomicAdd(&lh[digit], 1u);
    }
    __syncthreads();
    for (int i = threadIdx.x; i < 256; i += blockDim.x)
        if (lh[i]) atomicAdd(&hist[b * 256 + i], lh[i]);
}

// Scan: pick the digit containing the need-th largest candidate.
// WMMA path: 16 group-of-16 sums computed by 4 accumulating
// v_wmma_f32_16x16x4_f32 (B = all-ones -> D(m,*) = sum of row m of A;
// integer counts < 2^24 so f32 is exact), then the scan thread skips whole
// groups and walks only the 16 bins of the target group.
__global__ void k_scan(unsigned* hist, unsigned long long* pref,
                       unsigned* needArr, int round) {
    int b = blockIdx.x;
#if HAVE_WMMA4
    __shared__ float sgs[16];
    if (threadIdx.x < 32) {                          // wave 0, EXEC all ones
        const unsigned* h = hist + b * 256;
        int lane = threadIdx.x;
        int m  = lane & 15;                          // A-matrix row (doc layout)
        int kh = lane >> 4;                          // 0: K=0,1  1: K=2,3
        v8f c = {0.f, 0.f, 0.f, 0.f, 0.f, 0.f, 0.f, 0.f};
        v2f ones; ones.x = 1.0f; ones.y = 1.0f;      // B == ones, layout-free
#pragma unroll
        for (int s = 0; s < 4; ++s) {                // A_s(m,k) = h[m*16+s*4+k]
            v2f a;
            a.x = (float)h[m * 16 + s * 4 + 2 * kh + 0];
            a.y = (float)h[m * 16 + s * 4 + 2 * kh + 1];
            c = __builtin_amdgcn_wmma_f32_16x16x4_f32(false, a, false, ones,
                                                      (short)0, c, false, false);
        }
        // D layout: lane0 col0 -> rows 0..7 in c[0..7]; lane16 -> rows 8..15
        if (lane == 0) {
#pragma unroll
            for (int r = 0; r < 8; ++r) sgs[r] = c[r];
        } else if (lane == 16) {
#pragma unroll
            for (int r = 0; r < 8; ++r) sgs[8 + r] = c[r];
        }
        if (lane == 0) {
            unsigned need = needArr[b];
            int v = 0;
            for (int g = 15; g >= 0; --g) {
                unsigned gs = (unsigned)sgs[g];
                if (g > 0 && gs < need) { need -= gs; continue; }
                v = g * 16 + 15;
                for (; v > g * 16; --v) {
                    unsigned cc = h[v];
                    if (cc >= need) break;
                    need -= cc;
                }
                break;
            }
            pref[b] |= ((unsigned long long)(unsigned)v) << (56 - 8 * round);
            needArr[b] = need;
        }
    }
#else
    if (threadIdx.x == 0) {
        unsigned need = needArr[b];
        const unsigned* h = hist + b * 256;
        int v = 255;
        for (; v > 0; --v) {
            unsigned c = h[v];
            if (c >= need) break;
            need -= c;
        }
        pref[b] |= ((unsigned long long)(unsigned)v) << (56 - 8 * round);
        needArr[b] = need;
    }
#endif
    __syncthreads();                                  // readers done before reset
    for (int i = threadIdx.x; i < 256; i += blockDim.x) hist[b * 256 + i] = 0u;
}

// ============================================================================
// Kernel 3: compact the exactly-6000 keys >= T[b]
// ============================================================================
__global__ void k_compact(const unsigned long long* __restrict__ keys,
                          const unsigned long long* __restrict__ pref,
                          unsigned* __restrict__ cnt,
                          unsigned long long* __restrict__ topk) {
    int t = blockIdx.x * blockDim.x + threadIdx.x;
    int b = t / NPROP;
    unsigned long long key = keys[t];
    if (key >= pref[b]) {
        unsigned p = atomicAdd(&cnt[b], 1u);
        if (p < PRE_NMS) topk[(size_t)b * PRE_NMS + p] = key;
    }
}

// ============================================================================
// Kernel 4: per-batch sort of 6000 keys (bitonic, descending) in LDS.
// The 48 KB key block is DMA'd into LDS by the Tensor Data Mover.
// ============================================================================
__global__ __launch_bounds__(1024)
void k_sort(const unsigned long long* __restrict__ topk,
            const float4* __restrict__ prop,
            float4* __restrict__ sbox) {
    __shared__ unsigned long long sk[SORT_N];        // 64 KB
    int b = blockIdx.x;
    const unsigned long long* src = topk + (size_t)b * PRE_NMS;

#if HAVE_TDM
    if (threadIdx.x == 0) tdm_load_1d_u64(src, &sk[0], PRE_NMS);
    for (int t = PRE_NMS + threadIdx.x; t < SORT_N; t += blockDim.x) sk[t] = 0ull;
#else
    for (int t = threadIdx.x; t < SORT_N; t += blockDim.x)
        sk[t] = (t < PRE_NMS) ? src[t] : 0ull;
#endif
    __syncthreads();

    // bitonic sort, descending
    for (unsigned k = 2; k <= SORT_N; k <<= 1) {
        for (unsigned j = k >> 1; j > 0; j >>= 1) {
            for (unsigned i = threadIdx.x; i < SORT_N; i += blockDim.x) {
                unsigned l = i ^ j;
                if (l > i) {
                    unsigned long long a = sk[i], c = sk[l];
                    bool up = ((i & k) == 0);
                    if ((up && a < c) || (!up && a > c)) { sk[i] = c; sk[l] = a; }
                }
            }
            __syncthreads();
        }
    }

    // gather boxes into score-sorted order
    for (int t = threadIdx.x; t < PRE_NMS; t += blockDim.x) {
        unsigned long long key = sk[t];
        unsigned idx = 0xFFFFFFFFu - (unsigned)(key & 0xFFFFFFFFull);
        float4 v = (idx < (unsigned)NPROP) ? prop[(size_t)b * NPROP + idx]
                                           : make_float4(0.f, 0.f, 0.f, 0.f);
        sbox[(size_t)b * PRE_NMS + t] = v;
    }
}

// ============================================================================
// Kernel 5: greedy NMS + emit [b, x1,y1,x2,y2] x 300 (zero rows keep batch col)
// Each thread register-caches its 6 owned boxes; the pivot box stream is
// TDM-staged into LDS in two 48 KB chunks and read via LDS broadcast.
// ============================================================================
__global__ __launch_bounds__(1024)
void k_nms_out(const float4* __restrict__ sbox, float* __restrict__ out) {
    int b = blockIdx.x;
    const float4* boxes = sbox + (size_t)b * PRE_NMS;
    __shared__ float4 schunk[NMS_CHUNK];             // 48 KB pivot-box stage
    __shared__ unsigned char keep[6144];
    __shared__ int s_cnt;

    float4 bx[6];
    float  ar[6];
#pragma unroll
    for (int m = 0; m < 6; ++m) {
        int j = threadIdx.x + m * 1024;
        if (j < PRE_NMS) {
            float4 v = boxes[j];
            bx[m] = v;
            ar[m] = (v.z - v.x + 1.0f) * (v.w - v.y + 1.0f);
            keep[j] = 1;
        } else if (j < 6144) {
            keep[j] = 0;
        }
    }
    __syncthreads();

    for (int half = 0; half < 2; ++half) {
        const int i0 = half * NMS_CHUNK;
        // stage pivot boxes [i0, i0+NMS_CHUNK) into LDS
#if HAVE_TDM
        if (threadIdx.x == 0)
            tdm_load_1d_u64(boxes + i0, &schunk[0], NMS_CHUNK * 2); // float4 = 2 x u64
#else
        for (int t = threadIdx.x; t < NMS_CHUNK; t += blockDim.x)
            schunk[t] = boxes[i0 + t];
#endif
        __syncthreads();

        for (int i = i0; i < i0 + NMS_CHUNK; ++i) {
            if (keep[i]) {                           // uniform across block
                float4 bi = schunk[i - i0];          // LDS broadcast read
                float  ai = (bi.z - bi.x + 1.0f) * (bi.w - bi.y + 1.0f);
#pragma unroll
                for (int m = 0; m < 6; ++m) {
                    int j = threadIdx.x + m * 1024;
                    if (j > i && j < PRE_NMS && keep[j]) {
                        float iw = fminf(bi.z, bx[m].z) - fmaxf(bi.x, bx[m].x) + 1.0f;
                        float ih = fminf(bi.w, bx[m].w) - fmaxf(bi.y, bx[m].y) + 1.0f;
                        float inter = fmaxf(iw, 0.0f) * fmaxf(ih, 0.0f);
                        float iou = inter / (ai + ar[m] - inter);
                        if (iou > NMS_TH) keep[j] = 0;
                    }
                }
            }
            __syncthreads();   // also fences LDS before next chunk's DMA overwrite
        }
    }

    if (threadIdx.x == 0) {
        int r = 0;
        for (int i = 0; i < PRE_NMS && r < POST_NMS; ++i) {
            if (keep[i]) {
                float4 bi = boxes[i];
                float* row = out + ((size_t)b * POST_NMS + r) * 5;
                row[0] = (float)b;
                row[1] = bi.x; row[2] = bi.y; row[3] = bi.z; row[4] = bi.w;
                ++r;
            }
        }
        s_cnt = r;
    }
    __syncthreads();
    for (int r = s_cnt + (int)threadIdx.x; r < POST_NMS; r += 1024) {
        float* row = out + ((size_t)b * POST_NMS + r) * 5;
        row[0] = (float)b;
        row[1] = 0.0f; row[2] = 0.0f; row[3] = 0.0f; row[4] = 0.0f;
    }
}

// ============================================================================
extern "C" void kernel_launch(void* const* d_in, const int* in_sizes, int n_in,
                              void* d_out, int out_size, void* d_ws, size_t ws_size,
                              hipStream_t stream) {
    (void)in_sizes; (void)n_in; (void)out_size; (void)ws_size;
    const float* scores_map = (const float*)d_in[0];
    const float* bbox_delt  = (const float*)d_in[1];
    const float* im_info    = (const float*)d_in[2];
    const float* anchors    = (const float*)d_in[3];
    float* out = (float*)d_out;

    char* ws = (char*)d_ws;
    float4*             prop = (float4*)(ws + OFF_PROP);
    unsigned long long* keys = (unsigned long long*)(ws + OFF_KEYS);
    unsigned long long* topk = (unsigned long long*)(ws + OFF_TOPK);
    float4*             sbox = (float4*)(ws + OFF_SBOX);
    unsigned*           hist = (unsigned*)(ws + OFF_HIST);
    unsigned long long* pref = (unsigned long long*)(ws + OFF_PREF);
    unsigned*           need = (unsigned*)(ws + OFF_NEED);
    unsigned*           cnt  = (unsigned*)(ws + OFF_CNT);

    const int nblk = (BATCH * NPROP) / 256;   // 9216

    k_init<<<BATCH, 256, 0, stream>>>(hist, pref, need, cnt);
    k_transform<<<nblk, 256, 0, stream>>>(scores_map, bbox_delt, im_info, anchors,
                                          prop, keys);
    for (int r = 0; r < 8; ++r) {
        k_hist<<<nblk, 256, 0, stream>>>(keys, pref, hist, r);
        k_scan<<<BATCH, 256, 0, stream>>>(hist, pref, need, r);
    }
    k_compact<<<nblk, 256, 0, stream>>>(keys, pref, cnt, topk);
    k_sort<<<BATCH, 1024, 0, stream>>>(topk, prop, sbox);
    k_nms_out<<<BATCH, 1024, 0, stream>>>(sbox, out);
}